// Sensory2Hippo_56805237457584
// MI455X (gfx1250) — compile-verified
//
#include <hip/hip_runtime.h>
#include <math.h>
#include <stdint.h>

// Sensory2Hippo: out[b, g*D + d] = sigmoid(wp) * relu(xf[b,d]-mean(xf)) / max(||relu(row-mean)||2, eps)
// B=8192, D=1024, G=8. Pure bandwidth problem (~320MB moved); no real GEMM
// (W_tile is kron(ones, I) == 8x tiling), so no WMMA — the CDNA5 win is the
// async global->LDS path + non-temporal B128 output stores.

#define NROWS 8192
#define DIM   1024
#define NG    8
#define EPSN  1e-12f

typedef float v4f __attribute__((ext_vector_type(4)));

// ---------------------------------------------------------------------------
// Kernel A: per-block partial sums of xf. 1024 blocks x 256 threads.
// Each block covers 8192 consecutive floats (2048 float4); each thread 8 float4.
// Regular (RT) loads so xf stays resident in L2 for kernel C.
// ---------------------------------------------------------------------------
__global__ __launch_bounds__(256) void k_partial_sums(const float* __restrict__ xf,
                                                      float* __restrict__ partials) {
    const int tid = threadIdx.x;
    const v4f* p = (const v4f*)xf;
    const size_t base = (size_t)blockIdx.x * 2048 + tid;
    float s = 0.0f;
#pragma unroll
    for (int i = 0; i < 8; ++i) {
        v4f v = p[base + (size_t)i * 256];
        s += v.x + v.y + v.z + v.w;
    }
    // wave32 butterfly (deterministic)
#pragma unroll
    for (int off = 16; off > 0; off >>= 1) s += __shfl_xor(s, off, 32);
    __shared__ float wsum[8];
    const int wave = tid >> 5;
    if ((tid & 31) == 0) wsum[wave] = s;
    __syncthreads();
    if (tid == 0) {
        float t = 0.0f;
#pragma unroll
        for (int w = 0; w < 8; ++w) t += wsum[w];
        partials[blockIdx.x] = t;
    }
}

// ---------------------------------------------------------------------------
// Kernel B: reduce 1024 partials -> mean; compute sigmoid(wp). 1 block x 256.
// scalars[0] = mean, scalars[1] = sigmoid(wp)
// ---------------------------------------------------------------------------
__global__ __launch_bounds__(256) void k_finalize_scalars(const float* __restrict__ partials,
                                                          const float* __restrict__ wp,
                                                          float* __restrict__ scalars) {
    const int tid = threadIdx.x;
    float s = 0.0f;
#pragma unroll
    for (int i = 0; i < 4; ++i) s += partials[tid + i * 256];
#pragma unroll
    for (int off = 16; off > 0; off >>= 1) s += __shfl_xor(s, off, 32);
    __shared__ float wsum[8];
    const int wave = tid >> 5;
    if ((tid & 31) == 0) wsum[wave] = s;
    __syncthreads();
    if (tid == 0) {
        float t = 0.0f;
#pragma unroll
        for (int w = 0; w < 8; ++w) t += wsum[w];
        scalars[0] = t * (1.0f / ((float)NROWS * (float)DIM));   // global mean
        scalars[1] = 1.0f / (1.0f + expf(-wp[0]));               // sigmoid(wp)
    }
}

// ---------------------------------------------------------------------------
// Kernel C: one block per row. Async-copy the 4KB row into LDS
// (global_load_async_to_lds_b128, one 16B transfer per lane), wait on
// ASYNCcnt, compute relu/norm, write 8 tiled copies with NT B128 stores.
// ---------------------------------------------------------------------------
__global__ __launch_bounds__(256) void k_rows(const float* __restrict__ xf,
                                              const float* __restrict__ scalars,
                                              float* __restrict__ out) {
    __shared__ float tile[DIM];          // 4KB row staging
    __shared__ float wss[8];

    const int tid  = threadIdx.x;
    const int row  = blockIdx.x;
    const float mean  = scalars[0];
    const float scale = scalars[1];

    // ---- async global -> LDS copy of this row (CDNA5 ASYNCcnt path) ----
    {
        // LDS byte offset: low 32 bits of the generic pointer to a __shared__
        // object are the group-segment-relative offset (aperture in high bits).
        uint32_t lds_addr = (uint32_t)(uintptr_t)(&tile[tid * 4]);
        uint32_t voff     = (uint32_t)(tid * 16);                 // byte offset in row
        const float* srow = xf + (size_t)row * DIM;               // uniform -> SGPR pair
        asm volatile("global_load_async_to_lds_b128 %0, %1, %2"
                     :
                     : "v"(lds_addr), "v"(voff), "s"(srow)
                     : "memory");
        asm volatile("s_wait_asynccnt 0" ::: "memory");
    }
    __syncthreads();

    // ---- relu(x - mean) and row sum-of-squares ----
    v4f v = ((const v4f*)tile)[tid];     // ds_load_b128
    v4f y;
    y.x = fmaxf(v.x - mean, 0.0f);
    y.y = fmaxf(v.y - mean, 0.0f);
    y.z = fmaxf(v.z - mean, 0.0f);
    y.w = fmaxf(v.w - mean, 0.0f);
    float ss = y.x * y.x + y.y * y.y + y.z * y.z + y.w * y.w;
#pragma unroll
    for (int off = 16; off > 0; off >>= 1) ss += __shfl_xor(ss, off, 32);
    const int wave = tid >> 5;
    if ((tid & 31) == 0) wss[wave] = ss;
    __syncthreads();
    float tot = 0.0f;
#pragma unroll
    for (int w = 0; w < 8; ++w) tot += wss[w];   // same fixed order in every lane

    const float inv = scale / fmaxf(sqrtf(tot), EPSN);
    v4f o;
    o.x = y.x * inv; o.y = y.y * inv; o.z = y.z * inv; o.w = y.w * inv;

    // ---- write 8 tiled copies, non-temporal B128 (256MB stream > L2) ----
    float* orow = out + (size_t)row * (NG * DIM) + (size_t)tid * 4;
#pragma unroll
    for (int g = 0; g < NG; ++g) {
        __builtin_nontemporal_store(o, (v4f*)(orow + g * DIM));
    }
}

// ---------------------------------------------------------------------------
extern "C" void kernel_launch(void* const* d_in, const int* in_sizes, int n_in,
                              void* d_out, int out_size, void* d_ws, size_t ws_size,
                              hipStream_t stream) {
    const float* xf = (const float*)d_in[0];   // [8192,1024] f32
    const float* wp = (const float*)d_in[1];   // [1] f32
    // d_in[2] = W_tile is structured (kron(ones(1,8), I)); not needed.
    float* out = (float*)d_out;                // [8192, 8192] f32

    float* ws       = (float*)d_ws;
    float* partials = ws;                      // 1024 floats
    float* scalars  = ws + 1024;               // 2 floats (mean, sigmoid(wp))

    k_partial_sums<<<1024, 256, 0, stream>>>(xf, partials);
    k_finalize_scalars<<<1, 256, 0, stream>>>(partials, wp, scalars);
    k_rows<<<NROWS, 256, 0, stream>>>(xf, scalars, out);
}